// rawGraphConv_33225867002500
// MI455X (gfx1250) — compile-verified
//
#include <hip/hip_runtime.h>

// ---------------------------------------------------------------------------
// GraphConv-style: out = deg[:,None] * (x @ W^T) + b, deg = 1 + out-degree.
// FP32 WMMA (v_wmma_f32_16x16x4_f32) GEMM; kernel is HBM-bound (~230MB @ 23.3TB/s).
// ---------------------------------------------------------------------------

typedef float v2f __attribute__((ext_vector_type(2)));
typedef float v8f __attribute__((ext_vector_type(8)));

#define GC_N_NODES 100000
#define GC_N_EDGES 3200000
#define GC_DIM     256           // D_IN == D_OUT == 256

// ---- deg init: deg[i] = 1.0f --------------------------------------------
__global__ void gc_init_deg(float* __restrict__ deg, int n) {
  int i = blockIdx.x * blockDim.x + threadIdx.x;
  if (i < n) deg[i] = 1.0f;
}

// ---- histogram over edge sources (float atomics, resolved in L2) --------
__global__ void gc_degree(const int* __restrict__ src, float* __restrict__ deg, int nE) {
  int i = blockIdx.x * blockDim.x + threadIdx.x;
  if (i < nE) atomicAdd(&deg[src[i]], 1.0f);
}

// ---- fused GEMM + row-scale + bias --------------------------------------
// One wave32 computes a 16-row x 64-col tile of out (4 WMMA accumulators),
// K looped 0..255 in steps of 4 using V_WMMA_F32_16X16X4_F32.
//
// Fragment layouts (ISA 7.12.2, 32-bit 16x4 A / 4x16 B):
//   A: lane L holds x[m0 + L%16][kk + 2*(L/16) + {0,1}]   -> one 8B load
//   B: lane L holds W[o  + L%16][kk + 2*(L/16) + {0,1}]   (B = W^T tile)
//   C/D: VGPR j <-> row m0 + j + 8*(L/16), col o + L%16
__global__ __launch_bounds__(256) void gc_gemm_wmma(
    const float* __restrict__ x,       // [100000][256]
    const float* __restrict__ Wm,      // [256][256], W[o][k]
    const float* __restrict__ bias,    // [256]
    const float* __restrict__ deg,     // [100000]
    float* __restrict__ out)           // [100000][256]
{
  const int lane = threadIdx.x & 31;
  const int wave = threadIdx.x >> 5;
  const int gw   = blockIdx.x * 8 + wave;   // 0 .. 24999

  const int mTile  = gw >> 2;               // 0 .. 6249
  const int nStrip = gw & 3;                // 0 .. 3 (64 cols each)
  const int m0 = mTile * 16;
  const int o0 = nStrip * 64;

  const int ll = lane & 15;                 // row (A) / col (B) within tile
  const int lh = lane >> 4;                 // 0 or 1 -> K sub-offset 0 / 2

  const float* aP  = x  + (size_t)(m0 + ll) * GC_DIM + 2 * lh;
  const float* bP0 = Wm + (size_t)(o0 +  0 + ll) * GC_DIM + 2 * lh;
  const float* bP1 = bP0 + 16 * GC_DIM;
  const float* bP2 = bP0 + 32 * GC_DIM;
  const float* bP3 = bP0 + 48 * GC_DIM;

  v8f acc0 = {}, acc1 = {}, acc2 = {}, acc3 = {};

#pragma unroll 4
  for (int kk = 0; kk < GC_DIM; kk += 4) {
    v2f a  = *(const v2f*)(aP  + kk);
    v2f f0 = *(const v2f*)(bP0 + kk);
    v2f f1 = *(const v2f*)(bP1 + kk);
    v2f f2 = *(const v2f*)(bP2 + kk);
    v2f f3 = *(const v2f*)(bP3 + kk);
    acc0 = __builtin_amdgcn_wmma_f32_16x16x4_f32(false, a, false, f0, (short)0, acc0, false, false);
    acc1 = __builtin_amdgcn_wmma_f32_16x16x4_f32(false, a, false, f1, (short)0, acc1, false, false);
    acc2 = __builtin_amdgcn_wmma_f32_16x16x4_f32(false, a, false, f2, (short)0, acc2, false, false);
    acc3 = __builtin_amdgcn_wmma_f32_16x16x4_f32(false, a, false, f3, (short)0, acc3, false, false);
  }

  // Epilogue: out[row][col] = deg[row] * acc + bias[col]
  const int rowBase = m0 + 8 * lh;          // rows rowBase .. rowBase+7
  float dg[8];
#pragma unroll
  for (int j = 0; j < 8; ++j) dg[j] = deg[rowBase + j];

  const v8f* accs[4] = { &acc0, &acc1, &acc2, &acc3 };
#pragma unroll
  for (int t = 0; t < 4; ++t) {
    const int col = o0 + t * 16 + ll;
    const float bb = bias[col];
    float* op = out + (size_t)rowBase * GC_DIM + col;
    v8f a = *accs[t];
#pragma unroll
    for (int j = 0; j < 8; ++j) {
      op[(size_t)j * GC_DIM] = dg[j] * a[j] + bb;
    }
  }
}

extern "C" void kernel_launch(void* const* d_in, const int* in_sizes, int n_in,
                              void* d_out, int out_size, void* d_ws, size_t ws_size,
                              hipStream_t stream) {
  const float* x    = (const float*)d_in[0];            // [100000*256]
  const int*   eidx = (const int*)d_in[1];              // [2*3200000], row 0 = sources
  const float* Wm   = (const float*)d_in[2];            // [256*256]
  const float* bias = (const float*)d_in[3];            // [256]
  float*       out  = (float*)d_out;                    // [100000*256]
  float*       deg  = (float*)d_ws;                     // 100000 floats = 400KB scratch

  (void)in_sizes; (void)n_in; (void)out_size; (void)ws_size;

  // 1) deg = 1.0
  gc_init_deg<<<(GC_N_NODES + 255) / 256, 256, 0, stream>>>(deg, GC_N_NODES);

  // 2) deg[src] += 1 per edge
  gc_degree<<<(GC_N_EDGES + 255) / 256, 256, 0, stream>>>(eidx, deg, GC_N_EDGES);

  // 3) fused WMMA GEMM + scale + bias
  //    25000 waves total: 6250 m-tiles x 4 n-strips; 8 waves/block -> 3125 blocks
  gc_gemm_wmma<<<3125, 256, 0, stream>>>(x, Wm, bias, deg, out);
}